// Embedding_40252433498312
// MI455X (gfx1250) — compile-verified
//
#include <hip/hip_runtime.h>

#define CLASS_NUM   64
#define E_DIMS      256
#define TABLE_ELEMS (CLASS_NUM * E_DIMS)   // 16384 floats = 64 KB

typedef unsigned int u32x4 __attribute__((ext_vector_type(4)));
typedef int          i32x4 __attribute__((ext_vector_type(4)));
typedef int          i32x8 __attribute__((ext_vector_type(8)));
typedef float        f32x4 __attribute__((ext_vector_type(4)));

__device__ __forceinline__ float u32_as_f32(unsigned int v) {
    union { unsigned int u; float f; } c; c.u = v; return c.f;
}

// Stage the 64KB E_class table into LDS with the Tensor Data Mover.
// D# per CDNA5 ISA ch.8: group0 = {count/type, lds_addr, global_addr},
// group1 = {data_size, tensor/tile dims, strides}. One row of 16384 fp32.
__device__ __forceinline__ void stage_table_lds(const float* __restrict__ g,
                                                float* lds) {
#if __has_builtin(__builtin_amdgcn_tensor_load_to_lds)
    if (threadIdx.x < 32) {                      // one wave issues the DMA
        unsigned long long ga = (unsigned long long)(uintptr_t)g;
        unsigned int glo = (unsigned int)ga;
        unsigned int ghi = (unsigned int)(ga >> 32) & 0x1FFFFFFu; // addr[56:32]
        // generic pointer to LDS: low 32 bits are the LDS byte offset
        unsigned int lds_addr = (unsigned int)(uintptr_t)lds;

        u32x4 g0 = { 1u,                          // count=1, is_restore=0, gather=0
                     lds_addr,                    // lds_addr [63:32]
                     glo,                         // global_addr lo
                     ghi | 0x80000000u };         // global_addr hi | type=2 ([127:126])

        i32x8 g1 = { (int)0x00020000,             // data_size=2 (4B), mask=0
                     (int)0x40000000,             // tensor_dim0[15:0]=16384 @ [63:48]
                     (int)0x00010000,             // tensor_dim1=1 @ [95:80]
                     (int)0x40000000,             // tile_dim0=16384 @ [127:112]
                     (int)0x00000001,             // tile_dim1=1, tile_dim2=0
                     (int)0x00004000,             // tensor_dim0_stride=16384
                     0, 0 };                      // tensor_dim1_stride=0 (unused)
        i32x4 z4 = {0, 0, 0, 0};
#if __has_include(<hip/amd_detail/amd_gfx1250_TDM.h>)
        i32x8 z8 = {0, 0, 0, 0, 0, 0, 0, 0};
        __builtin_amdgcn_tensor_load_to_lds(g0, g1, z4, z4, z8, 0);
#else
        __builtin_amdgcn_tensor_load_to_lds(g0, g1, z4, z4, 0);
#endif
#if __has_builtin(__builtin_amdgcn_s_wait_tensorcnt)
        __builtin_amdgcn_s_wait_tensorcnt(0);
#else
        asm volatile("s_wait_tensorcnt 0" ::: "memory");
#endif
    }
#else
    // Fallback: cooperative vector copy
    for (int i = threadIdx.x; i < TABLE_ELEMS / 4; i += blockDim.x)
        ((f32x4*)lds)[i] = ((const f32x4*)g)[i];
#endif
    __syncthreads();   // all waves see staged table (issuer arrives post-wait)
}

__global__ void __launch_bounds__(256)
posenc_embed_kernel(const float* __restrict__ x,
                    const float* __restrict__ E_class,
                    float* __restrict__ out,
                    int n_tokens) {
    __shared__ float sE[TABLE_ELEMS];            // 64 KB of the 320 KB/WGP LDS
    stage_table_lds(E_class, sE);

    const int lane = threadIdx.x & 31;
    const int gw   = (int)((blockIdx.x * blockDim.x + threadIdx.x) >> 5);
    const int nw   = (int)((gridDim.x * blockDim.x) >> 5);

    for (int t = gw; t < n_tokens; t += nw) {    // one wave per token
        const float xv = x[t];
        f32x4* o = (f32x4*)(out + (size_t)t * E_DIMS) + lane * 2;

        // S=1024 is a multiple of 8, so (s % 8 == 0) <=> (t % 8 == 0).
        // t is wave-uniform, so this branch never diverges within a wave.
        if ((t & 7) == 0) {
            int idx = (int)xv;                   // trunc toward zero, x >= 0
            idx = idx < 0 ? 0 : (idx > CLASS_NUM - 1 ? CLASS_NUM - 1 : idx);
            const f32x4* src = (const f32x4*)(sE + idx * E_DIMS) + lane * 2;
            f32x4 a = src[0];
            f32x4 b = src[1];
            __builtin_nontemporal_store(a, &o[0]);   // streaming output: TH=NT
            __builtin_nontemporal_store(b, &o[1]);
        } else {
            // lane j covers k = 4j..4j+3; out[2k]=sin(pi*x*2^k), out[2k+1]=cos.
            // sin(pi*x*2^k) = sin(2*pi * x*2^(k-1)); hw v_sin/v_cos take turns,
            // so u = fract(x * 2^(k-1)) is an exact, in-range argument.
            const int   k0 = lane << 2;
            const float s0 = u32_as_f32((unsigned)(k0 + 126) << 23); // 2^(k0-1)
            const float u0 = __builtin_amdgcn_fractf(xv * s0);
            const float u1 = __builtin_amdgcn_fractf(xv * (s0 * 2.0f));
            const float u2 = __builtin_amdgcn_fractf(xv * (s0 * 4.0f));
            const float u3 = __builtin_amdgcn_fractf(xv * (s0 * 8.0f));
            f32x4 lo, hi;
            lo.x = __builtin_amdgcn_sinf(u0);  lo.y = __builtin_amdgcn_cosf(u0);
            lo.z = __builtin_amdgcn_sinf(u1);  lo.w = __builtin_amdgcn_cosf(u1);
            hi.x = __builtin_amdgcn_sinf(u2);  hi.y = __builtin_amdgcn_cosf(u2);
            hi.z = __builtin_amdgcn_sinf(u3);  hi.w = __builtin_amdgcn_cosf(u3);
            __builtin_nontemporal_store(lo, &o[0]);  // streaming output: TH=NT
            __builtin_nontemporal_store(hi, &o[1]);
        }
    }
}

extern "C" void kernel_launch(void* const* d_in, const int* in_sizes, int n_in,
                              void* d_out, int out_size, void* d_ws, size_t ws_size,
                              hipStream_t stream) {
    const float* x  = (const float*)d_in[0];   // [128,1024] fp32, values in [0,64)
    const float* Ec = (const float*)d_in[1];   // [64,256] fp32
    float* out      = (float*)d_out;           // [128,1024,256] fp32

    const int n_tokens = in_sizes[0];          // 131072
    const int threads  = 256;                  // 8 waves (wave32)
    const int blocks   = 2048;                 // 16384 waves -> 8 tokens/wave

    posenc_embed_kernel<<<blocks, threads, 0, stream>>>(x, Ec, out, n_tokens);
}